// STConvBlock_86199993631183
// MI455X (gfx1250) — compile-verified
//
#include <hip/hip_runtime.h>
#include <math.h>
#include <stdint.h>

// ---------------- problem constants ----------------
#define B_   8
#define T_   32
#define N_   500
#define F_   64
#define H_   128
#define E_   8000
#define KT_  3
#define BT_  (B_ * T_)        // 256
#define ROWS_ (BT_ * N_)      // 128000 (== 8000 * 16, exact M-tiles)
#define KEFF_ (KT_ * H_)      // 384
#define NTILES_ 8             // 128 / 16

typedef __attribute__((ext_vector_type(2))) float v2f;
typedef __attribute__((ext_vector_type(8))) float v8f;

__device__ __forceinline__ v8f wmma_f32x4(v2f a, v2f b, v8f c) {
    // V_WMMA_F32_16X16X4_F32: D = A(16x4,f32) * B(4x16,f32) + C(16x16,f32)
    return __builtin_amdgcn_wmma_f32_16x16x4_f32(
        /*neg_a=*/false, a, /*neg_b=*/false, b,
        /*c_mod=*/(short)0, c, /*reuse_a=*/false, /*reuse_b=*/false);
}

__device__ __forceinline__ float gelu_exact(float x) {
    return 0.5f * x * (1.0f + erff(x * 0.7071067811865476f));
}

// ---- CDNA5 async memory->LDS copy (GLOBAL_LOAD_ASYNC_TO_LDS_B128, ASYNCcnt) ----
__device__ __forceinline__ void async_ld_b128(void* lds_ptr, const void* g_ptr) {
    // low 32 bits of a generic LDS address are the DS byte address
    uint32_t lds_off = (uint32_t)(uintptr_t)lds_ptr;
    asm volatile("global_load_async_to_lds_b128 %0, %1, off"
                 :: "v"(lds_off), "v"(g_ptr)
                 : "memory");
}
__device__ __forceinline__ void wait_async0() {
    asm volatile("s_wait_asynccnt 0x0" ::: "memory");
}

// ---------------- graph preprocessing (tiny) ----------------
__global__ void k_graph_init(float* deg, int* counts, int* fill) {
    int i = blockIdx.x * blockDim.x + threadIdx.x;
    if (i < N_) { deg[i] = 1.0f; counts[i] = 0; fill[i] = 0; }  // self-loop weight 1
}

__global__ void k_deg_count(const int* __restrict__ ei, const float* __restrict__ ew,
                            float* deg, int* counts) {
    int e = blockIdx.x * blockDim.x + threadIdx.x;
    if (e < E_) {
        int d = ei[E_ + e];                 // edge_index[1][e] (dst)
        atomicAdd(&deg[d], ew[e]);
        atomicAdd(&counts[d], 1);
    }
}

__global__ void k_dinv(const float* __restrict__ deg, float* __restrict__ dinv) {
    int i = blockIdx.x * blockDim.x + threadIdx.x;
    if (i < N_) dinv[i] = deg[i] > 0.0f ? rsqrtf(deg[i]) : 0.0f;
}

__global__ void k_scan(const int* __restrict__ counts, int* __restrict__ rowstart) {
    if (blockIdx.x == 0 && threadIdx.x == 0) {
        int acc = 0; rowstart[0] = 0;
        for (int i = 0; i < N_; ++i) { acc += counts[i]; rowstart[i + 1] = acc; }
    }
}

__global__ void k_fill(const int* __restrict__ ei, const float* __restrict__ ew,
                       const float* __restrict__ dinv, const int* __restrict__ rowstart,
                       int* fill, int* __restrict__ cols, float* __restrict__ vals) {
    int e = blockIdx.x * blockDim.x + threadIdx.x;
    if (e < E_) {
        int s = ei[e], d = ei[E_ + e];
        int pos = rowstart[d] + atomicAdd(&fill[d], 1);
        cols[pos] = s;
        vals[pos] = dinv[s] * ew[e] * dinv[d];
    }
}

// ---------------- weight packing into WMMA B-fragment order ----------------
// dst[((kk/4)*NTILES + nt)*64 + l*2 + j] = B[kk + kb + j][nt*16 + (l&15)], kb=(l>>4)*2
__global__ void k_pack_kxn(const float* __restrict__ src, float* __restrict__ dst,
                           int ksteps) {
    int idx = blockIdx.x * blockDim.x + threadIdx.x;
    if (idx < ksteps * NTILES_ * 64) {
        int step = idx >> 9;            // /(8*64)
        int rem  = idx & 511;
        int nt   = rem >> 6;
        int q    = rem & 63;
        int l    = q >> 1, j = q & 1;
        int kb   = (l >> 4) << 1;
        int k    = step * 4 + kb + j;
        int n    = nt * 16 + (l & 15);
        dst[idx] = src[k * H_ + n];
    }
}

// fused transpose+pack for W_temp (O=128, I=128, K=3): logical B row = ktap*128 + i
__global__ void k_pack_wt(const float* __restrict__ W, float* __restrict__ dst) {
    int idx = blockIdx.x * blockDim.x + threadIdx.x;
    if (idx < (KEFF_ / 4) * NTILES_ * 64) {
        int step = idx >> 9;
        int rem  = idx & 511;
        int nt   = rem >> 6;
        int q    = rem & 63;
        int l    = q >> 1, j = q & 1;
        int kb   = (l >> 4) << 1;
        int kidx = step * 4 + kb + j;   // 0..383
        int ktap = kidx >> 7;
        int i    = kidx & 127;
        int o    = nt * 16 + (l & 15);
        dst[idx] = W[o * (H_ * KT_) + i * KT_ + ktap];
    }
}

// ---------------- xf = x @ W_gcn (fp32 WMMA) ----------------
__global__ void __launch_bounds__(256)
k_gcn_xf(const float* __restrict__ x, const float* __restrict__ Wgp,
         float* __restrict__ xf) {
    __shared__ float sx[16 * 68];             // 16x64 tile, padded stride 68 (bank rotate)
    const int mt  = blockIdx.x;               // 0..7999
    const int tid = threadIdx.x;

    {   // async-stage contiguous 16x64 A tile (one B128 async op per thread)
        const float* xa = x + (size_t)mt * 16 * F_ + tid * 4;
        int r = (tid * 4) >> 6, c = (tid * 4) & 63;
        async_ld_b128(&sx[r * 68 + c], xa);
        wait_async0();
    }
    __syncthreads();

    const int l    = tid & 31;
    const int nt   = tid >> 5;                // wave id = N-tile
    const int m    = l & 15;
    const int kb   = (l >> 4) << 1;           // K sub-offset {0,2}
    const int rh   = (l >> 4) << 3;           // C-row offset {0,8}
    const int ocol = nt * 16 + (l & 15);

    const float* Bp = Wgp + nt * 64 + l * 2;  // packed fragments, step stride 512
    v8f c = {};
    #pragma unroll
    for (int kk = 0; kk < F_; kk += 4) {
        v2f a = *(const v2f*)&sx[m * 68 + kk + kb];
        v2f b = *(const v2f*)(Bp + (kk >> 2) * 512);   // one coalesced b64 / lane
        c = wmma_f32x4(a, b, c);
    }

    const size_t r0 = (size_t)mt * 16;
    #pragma unroll
    for (int v = 0; v < 8; ++v)
        xf[(r0 + v + rh) * H_ + ocol] = c[v];
}

// ---------------- GCN aggregation + bias + gelu ----------------
__global__ void __launch_bounds__(256)
k_aggregate(const float* __restrict__ xf, const int* __restrict__ rowstart,
            const int* __restrict__ cols, const float* __restrict__ vals,
            const float* __restrict__ dinv, const float* __restrict__ bg,
            float* __restrict__ h) {
    int wid = blockIdx.x * 8 + (threadIdx.x >> 5);
    int l   = threadIdx.x & 31;
    int bt  = wid / N_;
    int n   = wid - bt * N_;
    int f   = l * 4;

    const float* base = xf + (size_t)bt * N_ * H_;
    float dn = dinv[n];
    float4 xs = *(const float4*)(base + (size_t)n * H_ + f);
    float4 acc;
    float sw = dn * dn;                        // self-loop norm
    acc.x = sw * xs.x; acc.y = sw * xs.y; acc.z = sw * xs.z; acc.w = sw * xs.w;

    int e1 = rowstart[n + 1];
    for (int e = rowstart[n]; e < e1; ++e) {
        int   s = cols[e];
        float w = vals[e];
        float4 xv = *(const float4*)(base + (size_t)s * H_ + f);
        acc.x += w * xv.x; acc.y += w * xv.y; acc.z += w * xv.z; acc.w += w * xv.w;
    }
    float4 bv = *(const float4*)(bg + f);
    float4 o;
    o.x = gelu_exact(acc.x + bv.x);
    o.y = gelu_exact(acc.y + bv.y);
    o.z = gelu_exact(acc.z + bv.z);
    o.w = gelu_exact(acc.w + bv.w);
    *(float4*)(h + ((size_t)bt * N_ + n) * H_ + f) = o;
}

// ---------------- temporal conv + gelu + residual(x@W_res+b) + LayerNorm ----------------
// one block per (b, node). conv as (32 x 384)@(384 x 128) WMMA GEMM with zero halo.
// Each wave owns ONE M-tile and TWO N-tiles: A-fragment loaded once, two WMMAs.
__global__ void __launch_bounds__(256)
k_temporal_ln(const float* __restrict__ h,  const float* __restrict__ x,
              const float* __restrict__ Wtp, const float* __restrict__ btmp,
              const float* __restrict__ Wrp, const float* __restrict__ br,
              const float* __restrict__ lnw, const float* __restrict__ lnb,
              float* __restrict__ out) {
    __shared__ float sh[34 * 132];   // (T + 2 halo) x H, padded stride (528B rows, 16B aligned)
    __shared__ float sx[32 * 68];    // T x F, padded stride (272B rows, 16B aligned)
    __shared__ float sy[32 * 132];   // T x H pre-LN
    __shared__ float psum[32 * 8], psq[32 * 8];
    __shared__ float smu[32], sinv[32];

    const int b    = blockIdx.x / N_;
    const int node = blockIdx.x - b * N_;
    const int tid  = threadIdx.x;

    {   // async-stage h sequence: row t = tid/8, 16 cols (4x async B128) per thread
        int t = tid >> 3, c0 = (tid & 7) * 16;
        const float* src = h + (((size_t)(b * T_ + t)) * N_ + node) * H_ + c0;
        float* d = &sh[(t + 1) * 132 + c0];
        async_ld_b128(d + 0,  src + 0);
        async_ld_b128(d + 4,  src + 4);
        async_ld_b128(d + 8,  src + 8);
        async_ld_b128(d + 12, src + 12);
    }
    {   // async-stage x sequence: row t = tid/8, 8 cols (2x async B128) per thread
        int t = tid >> 3, c0 = (tid & 7) * 8;
        const float* src = x + (((size_t)(b * T_ + t)) * N_ + node) * F_ + c0;
        float* d = &sx[t * 68 + c0];
        async_ld_b128(d + 0, src + 0);
        async_ld_b128(d + 4, src + 4);
    }

    // while the async engine fills LDS: warm near caches with packed conv weights
    #pragma unroll
    for (int p = 0; p < 3; ++p)
        __builtin_prefetch(Wtp + (tid + p * 256) * 64, 0, 0);

    if (tid < 132) { sh[tid] = 0.0f; sh[33 * 132 + tid] = 0.0f; }  // halo rows (DS path)

    wait_async0();
    __syncthreads();

    const int wave = tid >> 5, l = tid & 31;
    const int m   = l & 15;
    const int kb  = (l >> 4) << 1;
    const int rh  = (l >> 4) << 3;
    const int mt  = wave >> 2;                 // 0..1
    const int ntp = (wave & 3) * 2;            // N-tile pair {ntp, ntp+1}
    const int ocol0 = ntp * 16 + (l & 15);
    const int trow  = mt * 16 + m;

    // conv GEMM: K = 384, A-fragment shared between the two N-tiles
    const float* Bc = Wtp + ntp * 64 + l * 2;  // packed; +64 => next N-tile
    v8f c0 = {}, c1 = {};
    #pragma unroll 4
    for (int kk = 0; kk < KEFF_; kk += 4) {
        int k  = kk >> 7;                      // conv tap (uniform within 4-step)
        int i0 = kk & 127;
        v2f a  = *(const v2f*)&sh[(trow + k) * 132 + i0 + kb];  // t + k - 1 (+1 halo)
        const float* bp = Bc + (kk >> 2) * 512;
        v2f b0 = *(const v2f*)(bp);
        v2f b1 = *(const v2f*)(bp + 64);
        c0 = wmma_f32x4(a, b0, c0);
        c1 = wmma_f32x4(a, b1, c1);
    }

    // residual GEMM: K = 64
    const float* Br = Wrp + ntp * 64 + l * 2;
    v8f r0 = {}, r1 = {};
    #pragma unroll
    for (int kk = 0; kk < F_; kk += 4) {
        v2f a  = *(const v2f*)&sx[trow * 68 + kk + kb];
        const float* bp = Br + (kk >> 2) * 512;
        v2f b0 = *(const v2f*)(bp);
        v2f b1 = *(const v2f*)(bp + 64);
        r0 = wmma_f32x4(a, b0, r0);
        r1 = wmma_f32x4(a, b1, r1);
    }

    {   // epilogue: bias + gelu + residual -> sy
        float bt0 = btmp[ocol0], br0 = br[ocol0];
        float bt1 = btmp[ocol0 + 16], br1 = br[ocol0 + 16];
        #pragma unroll
        for (int v = 0; v < 8; ++v) {
            int tr = mt * 16 + v + rh;
            sy[tr * 132 + ocol0]      = gelu_exact(c0[v] + bt0) + (r0[v] + br0);
            sy[tr * 132 + ocol0 + 16] = gelu_exact(c1[v] + bt1) + (r1[v] + br1);
        }
    }
    __syncthreads();

    // LayerNorm over H=128 per time row: partial sums (8 segs of 16 per row)
    {
        int row = tid & 31, seg = tid >> 5;
        const float* yr = &sy[row * 132 + seg * 16];
        float s = 0.0f, q = 0.0f;
        #pragma unroll
        for (int j = 0; j < 16; ++j) { float v = yr[j]; s += v; q += v * v; }
        psum[row * 8 + seg] = s; psq[row * 8 + seg] = q;
    }
    __syncthreads();
    if (tid < 32) {
        float s = 0.0f, q = 0.0f;
        #pragma unroll
        for (int j = 0; j < 8; ++j) { s += psum[tid * 8 + j]; q += psq[tid * 8 + j]; }
        float mu  = s * (1.0f / 128.0f);
        float var = q * (1.0f / 128.0f) - mu * mu;
        smu[tid] = mu; sinv[tid] = rsqrtf(var + 1e-5f);
    }
    __syncthreads();
    {
        int t = tid >> 3, c0 = (tid & 7) * 16;
        float mu = smu[t], inv = sinv[t];
        float* d = out + (((size_t)(b * T_ + t)) * N_ + node) * H_ + c0;
        #pragma unroll
        for (int j = 0; j < 16; j += 4) {
            float4 y = *(const float4*)&sy[t * 132 + c0 + j];
            float4 o;
            o.x = (y.x - mu) * inv * lnw[c0 + j + 0] + lnb[c0 + j + 0];
            o.y = (y.y - mu) * inv * lnw[c0 + j + 1] + lnb[c0 + j + 1];
            o.z = (y.z - mu) * inv * lnw[c0 + j + 2] + lnb[c0 + j + 2];
            o.w = (y.w - mu) * inv * lnw[c0 + j + 3] + lnb[c0 + j + 3];
            *(float4*)(d + j) = o;
        }
    }
}

// ---------------- host launch ----------------
extern "C" void kernel_launch(void* const* d_in, const int* in_sizes, int n_in,
                              void* d_out, int out_size, void* d_ws, size_t ws_size,
                              hipStream_t stream) {
    const float* x   = (const float*)d_in[0];
    const int*   ei  = (const int*)  d_in[1];   // edge_index (2,E)
    const float* ew  = (const float*)d_in[2];
    const float* Wg  = (const float*)d_in[3];   // (64,128)
    const float* bg  = (const float*)d_in[4];
    const float* Wtm = (const float*)d_in[5];   // (128,128,3)
    const float* btm = (const float*)d_in[6];
    const float* lnw = (const float*)d_in[7];
    const float* lnb = (const float*)d_in[8];
    const float* Wr  = (const float*)d_in[9];   // (64,128)
    const float* br  = (const float*)d_in[10];
    float* out = (float*)d_out;

    // workspace carve-up (256B aligned)
    char* ws = (char*)d_ws;
    size_t off = 0;
    auto alloc = [&](size_t bytes) { char* p = ws + off; off = (off + bytes + 255) & ~(size_t)255; return p; };
    float* xf   = (float*)alloc((size_t)ROWS_ * H_ * 4);           // 65.5 MB
    float* hbuf = (float*)alloc((size_t)ROWS_ * H_ * 4);           // 65.5 MB
    float* Wtp  = (float*)alloc((size_t)(KEFF_ / 4) * 512 * 4);    // 192 KB packed conv W
    float* Wgp  = (float*)alloc((size_t)(F_ / 4) * 512 * 4);       // 32 KB packed W_gcn
    float* Wrp  = (float*)alloc((size_t)(F_ / 4) * 512 * 4);       // 32 KB packed W_res
    float* deg  = (float*)alloc(N_ * 4);
    float* dinv = (float*)alloc(N_ * 4);
    int*   cnts = (int*)  alloc(N_ * 4);
    int*   rows = (int*)  alloc((N_ + 1) * 4);
    int*   fill = (int*)  alloc(N_ * 4);
    int*   cols = (int*)  alloc(E_ * 4);
    float* vals = (float*)alloc(E_ * 4);

    // graph prep + weight packing
    k_graph_init<<<(N_ + 255) / 256, 256, 0, stream>>>(deg, cnts, fill);
    k_deg_count <<<(E_ + 255) / 256, 256, 0, stream>>>(ei, ew, deg, cnts);
    k_dinv      <<<(N_ + 255) / 256, 256, 0, stream>>>(deg, dinv);
    k_scan      <<<1, 1, 0, stream>>>(cnts, rows);
    k_fill      <<<(E_ + 255) / 256, 256, 0, stream>>>(ei, ew, dinv, rows, fill, cols, vals);
    k_pack_wt   <<<((KEFF_ / 4) * 512 + 255) / 256, 256, 0, stream>>>(Wtm, Wtp);
    k_pack_kxn  <<<((F_ / 4) * 512 + 255) / 256, 256, 0, stream>>>(Wg, Wgp, F_ / 4);
    k_pack_kxn  <<<((F_ / 4) * 512 + 255) / 256, 256, 0, stream>>>(Wr, Wrp, F_ / 4);

    // pipeline
    k_gcn_xf     <<<ROWS_ / 16, 256, 0, stream>>>(x, Wgp, xf);
    k_aggregate  <<<(BT_ * N_) / 8, 256, 0, stream>>>(xf, rows, cols, vals, dinv, bg, hbuf);
    k_temporal_ln<<<B_ * N_, 256, 0, stream>>>(hbuf, x, Wtp, btm, Wrp, br, lnw, lnb, out);
}